// SelectiveRoPE_84774064488853
// MI455X (gfx1250) — compile-verified
//
#include <hip/hip_runtime.h>
#include <math.h>

// ---------------- problem constants (from reference) ----------------
#define BB 4
#define TT 4096
#define MODEL_DIM 2048
#define NUM_HEADS 16
#define HEAD_K_DIM 128
#define PE_DIM 64
#define PHI_RANK 16
#define DD 1024            // NUM_HEADS * PE_DIM
#define NCHUNK 32
#define CHUNK 128          // TT / NCHUNK
#define KSEG 8             // split-K factor for GEMM1
#define KSEG_LEN 256       // MODEL_DIM / KSEG  (== 2 head chunks)
#define LOG_THETA 13.122363377404328f   // ln(500000)

typedef __attribute__((ext_vector_type(2))) float v2f;
typedef __attribute__((ext_vector_type(8))) float v8f;

// ---------------------------------------------------------------
// K0: per-(b,t,head) inverse L2 norm over 128 contiguous floats.
// tid == (b*T + t)*16 + head ; a wave of 32 threads covers two full
// 2048-float rows -> line-granularity coalesced.
// ---------------------------------------------------------------
__global__ void k0_norm(const float* __restrict__ hs, float* __restrict__ invn) {
    int tid = blockIdx.x * blockDim.x + threadIdx.x;   // B*T*16 threads
    const float4* p = (const float4*)(hs + (size_t)tid * HEAD_K_DIM);
    float acc = 0.f;
#pragma unroll
    for (int i = 0; i < HEAD_K_DIM / 4; ++i) {
        float4 v = p[i];
        acc += v.x * v.x + v.y * v.y + v.z * v.z + v.w * v.w;
    }
    invn[tid] = rsqrtf(acc + 1e-6f);
}

// ---------------------------------------------------------------
// K1: split-K partials of r = normalize(x) @ W1.
// 1024 row-tiles (16 timesteps) x 8 K-segments = 8192 waves (~8x the
// SIMD count -> latency hiding). Each wave runs 64 WMMAs across two
// interleaved accumulators, halving the dependent D->C chain.
// A-fragment (ISA 7.12.2, 32-bit A 16x4):
//   lane<16 : M=lane,    v0=K+0, v1=K+1 ; lane>=16: M=lane-16, v0=K+2, v1=K+3
// B (4x16) mirrors; C/D: VGPR i -> M = i + 8*(lane>>4), N = lane&15.
// ---------------------------------------------------------------
__global__ void k1_gemm1_wmma(const float* __restrict__ x,
                              const float* __restrict__ invn,
                              const float* __restrict__ W1,
                              float* __restrict__ part) {
    int tid   = blockIdx.x * blockDim.x + threadIdx.x;
    int wave  = tid >> 5;
    int lane  = threadIdx.x & 31;
    int seg   = wave & (KSEG - 1);
    int tile  = wave >> 3;
    int mrow  = lane & 15;
    int khalf = lane >> 4;                 // 0 or 1
    int row0  = tile * 16;                 // first timestep row of this tile

    const float* xrow = x    + (size_t)(row0 + mrow) * MODEL_DIM;
    const float* nrow = invn + (size_t)(row0 + mrow) * NUM_HEADS;

    v8f acc0 = {}, acc1 = {};
    // this segment covers heads [seg*2, seg*2+1]
    for (int hh = 0; hh < 2; ++hh) {
        int h = seg * 2 + hh;
        float s = nrow[h];                 // inv-norm of this 128-wide head chunk
        int kbase = h * HEAD_K_DIM;
#pragma unroll 4
        for (int kk = 0; kk < HEAD_K_DIM; kk += 8) {
            int k0 = kbase + kk;
            v2f a0, b0, a1, b1;
            a0.x = xrow[k0 + 2 * khalf]         * s;
            a0.y = xrow[k0 + 2 * khalf + 1]     * s;
            b0.x = W1[(size_t)(k0 + 2 * khalf)     * PHI_RANK + mrow];
            b0.y = W1[(size_t)(k0 + 2 * khalf + 1) * PHI_RANK + mrow];
            a1.x = xrow[k0 + 4 + 2 * khalf]     * s;
            a1.y = xrow[k0 + 4 + 2 * khalf + 1] * s;
            b1.x = W1[(size_t)(k0 + 4 + 2 * khalf)     * PHI_RANK + mrow];
            b1.y = W1[(size_t)(k0 + 4 + 2 * khalf + 1) * PHI_RANK + mrow];
            acc0 = __builtin_amdgcn_wmma_f32_16x16x4_f32(
                false, a0, false, b0, (short)0, acc0, false, false);
            acc1 = __builtin_amdgcn_wmma_f32_16x16x4_f32(
                false, a1, false, b1, (short)0, acc1, false, false);
        }
    }
    v8f acc = acc0 + acc1;
    float* pseg = part + (size_t)seg * (BB * TT * PHI_RANK);
#pragma unroll
    for (int i = 0; i < 8; ++i) {
        int m = row0 + i + (khalf << 3);
        pseg[(size_t)m * PHI_RANK + mrow] = acc[i];
    }
}

// K1r: deterministic split-K reduction (fixed summation order, no atomics).
__global__ void k1r_reduce(const float* __restrict__ part, float* __restrict__ r) {
    int i = blockIdx.x * blockDim.x + threadIdx.x;     // B*T*16 threads
    float s = 0.f;
#pragma unroll
    for (int seg = 0; seg < KSEG; ++seg)
        s += part[(size_t)seg * (BB * TT * PHI_RANK) + i];
    r[i] = s;
}

// phi_raw(t, d) = dot16(r[b,t,:], W2[:,d]); W2 column lives in registers.
__device__ __forceinline__ float phi_raw_at(const float* __restrict__ rb,
                                            const float* __restrict__ wc, int t) {
    if (t < 0) return 0.f;                 // causal conv zero-padding
    const float* rr = rb + (size_t)t * PHI_RANK;
    float s = 0.f;
#pragma unroll
    for (int kk = 0; kk < PHI_RANK; ++kk) s = fmaf(rr[kk], wc[kk], s);
    return s;
}

// ---------------------------------------------------------------
// K2a: per-(b, chunk, d) sum of conv(phi_raw) over a 128-step chunk.
// b/chunk derive from blockIdx only -> r row addresses are wave-uniform
// (L2 broadcasts); W2 column / conv taps / q-k accesses are d-contiguous.
// ---------------------------------------------------------------
__global__ void k2a_chunksum(const float* __restrict__ r,
                             const float* __restrict__ W2,
                             const float* __restrict__ cw,
                             float* __restrict__ S) {
    int dblk  = blockIdx.x & 3;            // DD/256 = 4
    int rest  = blockIdx.x >> 2;
    int chunk = rest & (NCHUNK - 1);
    int b     = rest >> 5;
    int d     = dblk * 256 + threadIdx.x;

    float wc[PHI_RANK];
#pragma unroll
    for (int kk = 0; kk < PHI_RANK; ++kk) wc[kk] = W2[(size_t)kk * DD + d];
    float c0 = cw[d], c1 = cw[DD + d], c2 = cw[2 * DD + d], c3 = cw[3 * DD + d];

    const float* rb = r + (size_t)b * TT * PHI_RANK;
    int t0 = chunk * CHUNK;
    float w0 = phi_raw_at(rb, wc, t0 - 3);
    float w1 = phi_raw_at(rb, wc, t0 - 2);
    float w2 = phi_raw_at(rb, wc, t0 - 1);
    float acc = 0.f;
    for (int t = t0; t < t0 + CHUNK; ++t) {
        float w3 = phi_raw_at(rb, wc, t);
        acc += c0 * w0 + c1 * w1 + c2 * w2 + c3 * w3;
        w0 = w1; w1 = w2; w2 = w3;
    }
    S[((size_t)b * NCHUNK + chunk) * DD + d] = acc;
}

// K2b: exclusive scan of the 32 chunk sums per (b,d). In-place is safe:
// K2a fully rewrites S at the start of every kernel_launch call.
__global__ void k2b_scan(float* __restrict__ S) {
    int tid = blockIdx.x * blockDim.x + threadIdx.x;   // B*DD threads
    int b = tid >> 10, d = tid & (DD - 1);
    float run = 0.f;
    for (int c = 0; c < NCHUNK; ++c) {
        size_t idx = ((size_t)b * NCHUNK + c) * DD + d;
        float v = S[idx];
        S[idx] = run;
        run += v;
    }
}

// ---------------------------------------------------------------
// K2c: redo the chunk recurrence with the scanned prefix as the running
// cumsum -> phi_tilde, then apply the selective RoPE rotation to q and k.
// d = head*64 + j; both halves of the 128-dim head use angle[j].
// ---------------------------------------------------------------
__global__ void k2c_rope(const float* __restrict__ r,
                         const float* __restrict__ W2,
                         const float* __restrict__ cw,
                         const float* __restrict__ S,
                         const float* __restrict__ q,
                         const float* __restrict__ k,
                         float* __restrict__ oq,
                         float* __restrict__ ok) {
    int dblk  = blockIdx.x & 3;
    int rest  = blockIdx.x >> 2;
    int chunk = rest & (NCHUNK - 1);
    int b     = rest >> 5;
    int d     = dblk * 256 + threadIdx.x;
    int head  = d >> 6;
    int j     = d & (PE_DIM - 1);
    float temp = __expf(-(float)j * (LOG_THETA / 64.0f));   // THETA^(-j/64)

    float wc[PHI_RANK];
#pragma unroll
    for (int kk = 0; kk < PHI_RANK; ++kk) wc[kk] = W2[(size_t)kk * DD + d];
    float c0 = cw[d], c1 = cw[DD + d], c2 = cw[2 * DD + d], c3 = cw[3 * DD + d];

    const float* rb = r + (size_t)b * TT * PHI_RANK;
    int t0 = chunk * CHUNK;
    float run = S[((size_t)b * NCHUNK + chunk) * DD + d];   // exclusive prefix
    float w0 = phi_raw_at(rb, wc, t0 - 3);
    float w1 = phi_raw_at(rb, wc, t0 - 2);
    float w2 = phi_raw_at(rb, wc, t0 - 1);

    for (int t = t0; t < t0 + CHUNK; ++t) {
        float w3 = phi_raw_at(rb, wc, t);
        run += c0 * w0 + c1 * w1 + c2 * w2 + c3 * w3;        // phi_tilde
        w0 = w1; w1 = w2; w2 = w3;

        float sn, cs;
        __sincosf(temp * run, &sn, &cs);

        size_t base = (((size_t)b * TT + t) * NUM_HEADS + head) * HEAD_K_DIM + j;
        float q0 = q[base], q1 = q[base + PE_DIM];
        float k0 = k[base], k1 = k[base + PE_DIM];
        oq[base]          = q0 * cs - q1 * sn;
        oq[base + PE_DIM] = q0 * sn + q1 * cs;
        ok[base]          = k0 * cs - k1 * sn;
        ok[base + PE_DIM] = k0 * sn + k1 * cs;
    }
}

extern "C" void kernel_launch(void* const* d_in, const int* in_sizes, int n_in,
                              void* d_out, int out_size, void* d_ws, size_t ws_size,
                              hipStream_t stream) {
    (void)in_sizes; (void)n_in; (void)out_size; (void)ws_size;
    const float* q  = (const float*)d_in[0];
    const float* k  = (const float*)d_in[1];
    const float* hs = (const float*)d_in[2];
    const float* W1 = (const float*)d_in[3];
    const float* W2 = (const float*)d_in[4];
    const float* cw = (const float*)d_in[5];

    float* oq = (float*)d_out;
    float* ok = oq + (size_t)BB * TT * NUM_HEADS * HEAD_K_DIM;

    // workspace carve-out (~10.5 MB)
    float* invn = (float*)d_ws;                         // B*T*16 floats (1 MB)
    float* r    = invn + (size_t)BB * TT * NUM_HEADS;   // B*T*16 floats (1 MB)
    float* S    = r    + (size_t)BB * TT * PHI_RANK;    // B*NCHUNK*DD (512 KB)
    float* part = S    + (size_t)BB * NCHUNK * DD;      // KSEG * B*T*16 (8 MB)

    // K0: norms.  B*T*16 = 262144 threads
    k0_norm<<<(BB * TT * NUM_HEADS) / 256, 256, 0, stream>>>(hs, invn);

    // K1: WMMA split-K GEMM.  1024 tiles * 8 segs = 8192 waves
    k1_gemm1_wmma<<<(BB * TT / 16) * KSEG * 32 / 256, 256, 0, stream>>>(hs, invn, W1, part);

    // K1r: deterministic split-K reduction
    k1r_reduce<<<(BB * TT * PHI_RANK) / 256, 256, 0, stream>>>(part, r);

    // K2a: chunk partial sums.  B*NCHUNK*(DD/256) = 512 blocks
    k2a_chunksum<<<BB * NCHUNK * (DD / 256), 256, 0, stream>>>(r, W2, cw, S);

    // K2b: exclusive scan of chunk sums.  B*DD = 4096 threads
    k2b_scan<<<(BB * DD) / 256, 256, 0, stream>>>(S);

    // K2c: cumsum + rotation + writeback
    k2c_rope<<<BB * NCHUNK * (DD / 256), 256, 0, stream>>>(r, W2, cw, S, q, k, oq, ok);
}